// Model_40767829573946
// MI455X (gfx1250) — compile-verified
//
#include <hip/hip_runtime.h>
#include <hip/hip_bf16.h>
#include <cstdint>
#include <cstddef>

#define B_   8
#define N_   4096
#define C_   64
#define G_   1024
#define K_   32
#define CH_  67      // C_ + 3
#define OUTCH_ 131   // CH_ + C_
#define EPS_ 1e-5f

typedef __attribute__((ext_vector_type(2))) float v2f;
typedef __attribute__((ext_vector_type(8))) float v8f;

// ---------- wave32 helpers (CDNA5 is wave32-only) ----------
__device__ __forceinline__ void wave_argmax(float& v, int& ix) {
#pragma unroll
  for (int off = 16; off > 0; off >>= 1) {
    float ov = __shfl_xor(v, off, 32);
    int   oi = __shfl_xor(ix, off, 32);
    if (ov > v || (ov == v && oi < ix)) { v = ov; ix = oi; }
  }
}
__device__ __forceinline__ void wave_argmin(float& v, int& ix) {
#pragma unroll
  for (int off = 16; off > 0; off >>= 1) {
    float ov = __shfl_xor(v, off, 32);
    int   oi = __shfl_xor(ix, off, 32);
    if (ov < v || (ov == v && oi < ix)) { v = ov; ix = oi; }
  }
}
__device__ __forceinline__ float wave_sum(float v) {
#pragma unroll
  for (int off = 16; off > 0; off >>= 1) v += __shfl_xor(v, off, 32);
  return v;
}

// ---------------------------------------------------------------------------
// Kernel 1: furthest point sampling. One block per batch, 1024 threads.
// xyz kept resident in LDS; per-thread min-dist in registers.
// Matches jnp.argmax first-occurrence tie-break via (val, smaller-idx) compare.
// ---------------------------------------------------------------------------
__global__ __launch_bounds__(1024)
void fps_kernel(const float* __restrict__ xyz, int* __restrict__ fps_idx,
                float* __restrict__ new_xyz /* d_out[0 .. B*G*3) */) {
  __shared__ float s_xyz[N_ * 3];
  __shared__ float s_redv[32];
  __shared__ int   s_redi[32];
  __shared__ int   s_fps[G_];
  __shared__ int   s_far;

  const int b    = blockIdx.x;
  const int tid  = threadIdx.x;
  const int lane = tid & 31;
  const int wv   = tid >> 5;
  const float* xb = xyz + (size_t)b * N_ * 3;

  for (int i = tid; i < N_ * 3; i += 1024) s_xyz[i] = xb[i];
  __syncthreads();

  // 4 points per thread, contiguous so local scan keeps smallest index on ties
  float px[4], py[4], pz[4], md[4];
#pragma unroll
  for (int i = 0; i < 4; ++i) {
    int n = tid * 4 + i;
    px[i] = s_xyz[n * 3 + 0];
    py[i] = s_xyz[n * 3 + 1];
    pz[i] = s_xyz[n * 3 + 2];
    md[i] = 1e10f;
  }

  int far = 0;
  for (int s = 0; s < G_; ++s) {
    if (tid == 0) s_fps[s] = far;
    float cx = s_xyz[far * 3 + 0];
    float cy = s_xyz[far * 3 + 1];
    float cz = s_xyz[far * 3 + 2];
    float bv = -1.0f; int bi = 0;
#pragma unroll
    for (int i = 0; i < 4; ++i) {
      float dx = px[i] - cx, dy = py[i] - cy, dz = pz[i] - cz;
      float d = dx * dx + dy * dy + dz * dz;
      md[i] = fminf(md[i], d);
      if (md[i] > bv) { bv = md[i]; bi = tid * 4 + i; }
    }
    wave_argmax(bv, bi);
    if (lane == 0) { s_redv[wv] = bv; s_redi[wv] = bi; }
    __syncthreads();
    if (wv == 0) {
      float v = s_redv[lane]; int ix = s_redi[lane];
      wave_argmax(v, ix);
      if (lane == 0) s_far = ix;
    }
    __syncthreads();
    far = s_far;
  }
  __syncthreads();

  // gather sampled coords + publish indices (blockDim.x == G_)
  int id = s_fps[tid];
  fps_idx[b * G_ + tid] = id;
  new_xyz[(size_t)(b * G_ + tid) * 3 + 0] = s_xyz[id * 3 + 0];
  new_xyz[(size_t)(b * G_ + tid) * 3 + 1] = s_xyz[id * 3 + 1];
  new_xyz[(size_t)(b * G_ + tid) * 3 + 2] = s_xyz[id * 3 + 2];
}

// ---------------------------------------------------------------------------
// Kernel 2: pairwise sqrdists via V_WMMA_F32_16X16X4_F32 + stable top-K=32.
// 16 groups per block; dist tile 16x4096 f32 in 256KB dynamic LDS (CDNA5
// allows up to 320KB LDS per workgroup). 8 waves: phase A tiles the N axis
// with rank-4 WMMA, phase B does per-group iterative (val,idx) argmin.
//   d[m][n] = qq[m]*1 + (-2)*(q·p) + pp[n]  == A(16x4) x B(4x16) + C
// A layout (ISA 7.12.2, 32-bit A 16x4): lanes0-15 {K0,K1}, lanes16-31 {K2,K3}
// B layout (row-striped, K and K+2 across lane halves), C/D: M=j / M=j+8.
// ---------------------------------------------------------------------------
__global__ __launch_bounds__(256)
void knn_kernel(const float* __restrict__ xyz, const float* __restrict__ new_xyz,
                int* __restrict__ knn) {
  extern __shared__ float sdist[];          // [16][4096]
  const int b    = blockIdx.x >> 6;         // G_/16 == 64 tiles per batch
  const int g0   = (blockIdx.x & 63) << 4;
  const int tid  = threadIdx.x;
  const int lane = tid & 31;
  const int wv   = tid >> 5;
  const int m    = lane & 15;
  const bool lo  = (lane < 16);
  const float* xb = xyz + (size_t)b * N_ * 3;

  // A operand: row m = query (g0+m): (qx,qy,qz,qq)
  const float* q = new_xyz + (size_t)(b * G_ + g0 + m) * 3;
  float qx = q[0], qy = q[1], qz = q[2];
  float qq = qx * qx + qy * qy + qz * qz;
  v2f a;
  a.x = lo ? qx : qz;
  a.y = lo ? qy : qq;

  // phase A: 256 tiles of 16 columns, round-robin over 8 waves
  for (int t = wv; t < N_ / 16; t += 8) {
    int n = t * 16 + m;
    float pxv = xb[n * 3 + 0], pyv = xb[n * 3 + 1], pzv = xb[n * 3 + 2];
    float pp = pxv * pxv + pyv * pyv + pzv * pzv;
    v2f bb;
    bb.x = lo ? (-2.0f * pxv) : (-2.0f * pzv);   // K0 / K2
    bb.y = lo ? (-2.0f * pyv) : (1.0f);          // K1 / K3
    v8f c = {pp, pp, pp, pp, pp, pp, pp, pp};    // C[m][n] = pp[n]
    v8f d = __builtin_amdgcn_wmma_f32_16x16x4_f32(
        /*neg_a=*/false, a, /*neg_b=*/false, bb,
        /*c_mod=*/(short)0, c, /*reuse_a=*/false, /*reuse_b=*/false);
    int rbase = lo ? 0 : 8;
#pragma unroll
    for (int j = 0; j < 8; ++j)
      sdist[(rbase + j) * N_ + t * 16 + m] = d[j];
  }
  __syncthreads();

  // phase B: one wave per group (2 groups per wave). Lane-strided stripes
  // (n % 32 == lane) are bank-conflict free; lane-local cached min so only
  // the winning lane rescans its stripe each selection round.
  for (int gi = wv; gi < 16; gi += 8) {
    float* drow = sdist + gi * N_;
    float bv = 1e30f; int bi = 0x7fffffff;
    for (int i = 0; i < N_ / 32; ++i) {
      int n = i * 32 + lane;
      float v = drow[n];
      if (v < bv) { bv = v; bi = n; }            // strict <: keeps smallest n
    }
    int* krow = knn + (size_t)(b * G_ + g0 + gi) * K_;
    for (int r = 0; r < K_; ++r) {
      float wvv = bv; int wi = bi;
      wave_argmin(wvv, wi);
      if (lane == 0) krow[r] = wi;
      if (wi == bi) {                            // only owner lane matches
        drow[wi] = 1e30f;
        bv = 1e30f; bi = 0x7fffffff;
        for (int i = 0; i < N_ / 32; ++i) {
          int n = i * 32 + lane;
          float v = drow[n];
          if (v < bv) { bv = v; bi = n; }
        }
      }
    }
  }
}

// ---------------------------------------------------------------------------
// Kernel 3: per-group channel means + per-group partial sums for the
// batch-wide std. One wave per group; lane k == neighbor k (K_ == wave32).
// ---------------------------------------------------------------------------
__global__ __launch_bounds__(256)
void stats_kernel(const float* __restrict__ xyz, const float* __restrict__ points,
                  const int* __restrict__ knn, float* __restrict__ means,
                  float* __restrict__ partials) {
  const int lane = threadIdx.x & 31;
  const int wv   = threadIdx.x >> 5;
  const int gid  = blockIdx.x * 8 + wv;          // 0 .. B*G-1
  const int b    = gid / G_;
  const float* pb = points + (size_t)b * N_ * C_;
  const float* xb = xyz + (size_t)b * N_ * 3;
  const int nid = knn[(size_t)gid * K_ + lane];

  float sd = 0.0f, sdd = 0.0f;
  for (int c = 0; c < CH_; ++c) {
    float v = (c < C_) ? pb[(size_t)nid * C_ + c] : xb[(size_t)nid * 3 + (c - C_)];
    float mean = wave_sum(v) * (1.0f / (float)K_);
    if (lane == 0) means[(size_t)gid * CH_ + c] = mean;
    float d = v - mean;
    sd  += d;
    sdd += d * d;
  }
  sd  = wave_sum(sd);
  sdd = wave_sum(sdd);
  if (lane == 0) {
    partials[(size_t)gid * 2 + 0] = sd;
    partials[(size_t)gid * 2 + 1] = sdd;
  }
}

// ---------------------------------------------------------------------------
// Kernel 4: per-batch std -> scale = 1/(std+eps).
// sum((d-m2)^2) = sum(d^2) - n*m2^2  (m2 ~ 0 so no cancellation issue)
// ---------------------------------------------------------------------------
__global__ __launch_bounds__(256)
void reduce_kernel(const float* __restrict__ partials, float* __restrict__ scale) {
  __shared__ float ssd[256], ssdd[256];
  const int b = blockIdx.x, tid = threadIdx.x;
  float sd = 0.0f, sdd = 0.0f;
  for (int i = tid; i < G_; i += 256) {
    sd  += partials[(size_t)(b * G_ + i) * 2 + 0];
    sdd += partials[(size_t)(b * G_ + i) * 2 + 1];
  }
  ssd[tid] = sd; ssdd[tid] = sdd;
  __syncthreads();
  for (int s = 128; s > 0; s >>= 1) {
    if (tid < s) { ssd[tid] += ssd[tid + s]; ssdd[tid] += ssdd[tid + s]; }
    __syncthreads();
  }
  if (tid == 0) {
    const float n = (float)(G_ * K_ * CH_);      // 2,195,456 (exact in f32)
    float m2  = ssd[0] / n;
    float num = ssdd[0] - n * m2 * m2;
    float var = fmaxf(num, 0.0f) / (n - 1.0f);
    scale[b] = 1.0f / (sqrtf(var) + EPS_);
  }
}

// ---------------------------------------------------------------------------
// Kernel 5: gather + normalize + affine + concat center features.
// One block per group; streaming, mostly-coalesced 131-channel rows.
// ---------------------------------------------------------------------------
__global__ __launch_bounds__(256)
void finalize_kernel(const float* __restrict__ xyz, const float* __restrict__ points,
                     const float* __restrict__ alpha, const float* __restrict__ beta,
                     const int* __restrict__ fps_idx, const int* __restrict__ knn,
                     const float* __restrict__ means, const float* __restrict__ scale,
                     float* __restrict__ out_main) {
  __shared__ int   s_knn[K_];
  __shared__ float s_mean[CH_];
  __shared__ float s_alpha[CH_], s_beta[CH_];
  __shared__ int   s_cidx;
  __shared__ float s_scale;

  const int gid = blockIdx.x;                    // b*G + g
  const int b   = gid / G_;
  const int tid = threadIdx.x;
  if (tid < K_)  s_knn[tid]  = knn[(size_t)gid * K_ + tid];
  if (tid < CH_) {
    s_mean[tid]  = means[(size_t)gid * CH_ + tid];
    s_alpha[tid] = alpha[tid];
    s_beta[tid]  = beta[tid];
  }
  if (tid == 128) s_cidx  = fps_idx[gid];
  if (tid == 129) s_scale = scale[b];
  __syncthreads();

  const float* pb = points + (size_t)b * N_ * C_;
  const float* xb = xyz + (size_t)b * N_ * 3;
  float* orow = out_main + (size_t)gid * K_ * OUTCH_;

  for (int e = tid; e < K_ * OUTCH_; e += 256) {
    int k  = e / OUTCH_;
    int ch = e - k * OUTCH_;
    float val;
    if (ch < CH_) {
      int nid = s_knn[k];
      float v = (ch < C_) ? pb[(size_t)nid * C_ + ch] : xb[(size_t)nid * 3 + (ch - C_)];
      val = s_alpha[ch] * ((v - s_mean[ch]) * s_scale) + s_beta[ch];
    } else {
      val = pb[(size_t)s_cidx * C_ + (ch - CH_)];
    }
    orow[e] = val;
  }
}

// ---------------------------------------------------------------------------
extern "C" void kernel_launch(void* const* d_in, const int* in_sizes, int n_in,
                              void* d_out, int out_size, void* d_ws, size_t ws_size,
                              hipStream_t stream) {
  (void)in_sizes; (void)n_in; (void)out_size; (void)ws_size;
  const float* xyz    = (const float*)d_in[0];   // [8,4096,3]
  const float* points = (const float*)d_in[1];   // [8,4096,64]
  const float* alpha  = (const float*)d_in[2];   // [67]
  const float* beta   = (const float*)d_in[3];   // [67]
  float* out = (float*)d_out;                    // new_xyz (B*G*3) ++ out (B*G*K*131)

  // workspace layout
  int*   fpsi     = (int*)d_ws;                          // B*G
  int*   knn      = fpsi + B_ * G_;                      // B*G*K
  float* means    = (float*)(knn + B_ * G_ * K_);        // B*G*67
  float* partials = means + (size_t)B_ * G_ * CH_;       // B*G*2
  float* scalev   = partials + (size_t)B_ * G_ * 2;      // B

  float* new_xyz  = out;                                 // output 0
  float* out_main = out + (size_t)B_ * G_ * 3;           // output 1

  fps_kernel<<<B_, 1024, 0, stream>>>(xyz, fpsi, new_xyz);
  knn_kernel<<<B_ * (G_ / 16), 256, 16 * N_ * sizeof(float), stream>>>(xyz, new_xyz, knn);
  stats_kernel<<<B_ * G_ / 8, 256, 0, stream>>>(xyz, points, knn, means, partials);
  reduce_kernel<<<B_, 256, 0, stream>>>(partials, scalev);
  finalize_kernel<<<B_ * G_, 256, 0, stream>>>(xyz, points, alpha, beta,
                                               fpsi, knn, means, scalev, out_main);
}